// BatchConv2D__34909494182351
// MI455X (gfx1250) — compile-verified
//
#include <hip/hip_runtime.h>

typedef float v2f __attribute__((ext_vector_type(2)));
typedef float v8f __attribute__((ext_vector_type(8)));
typedef unsigned int v4u __attribute__((ext_vector_type(4)));
typedef int v4i __attribute__((ext_vector_type(4)));
typedef int v8i __attribute__((ext_vector_type(8)));

#define CIN   32
#define COUT  64
#define Hh    128
#define Ww    128
#define HO    126
#define WOo   126
#define HW    (Hh*Ww)
#define HOWO  (HO*WOo)
#define KTOT  288          // CIN*KH*KW
#define KPAD  292          // padded row stride (bank-conflict-free, mult of 4)
#define NKD   (COUT*KPAD)  // 18,688 floats = 74,752 B
#define MAXCN 64

#if defined(__has_builtin)
#if __has_builtin(__builtin_amdgcn_tensor_load_to_lds) && \
    __has_builtin(__builtin_amdgcn_s_wait_tensorcnt)
#define HAVE_TDM 1
#endif
#endif

// ---------------------------------------------------------------------------
// Fold the padded kernel bank into a dense kernel, laid out as
// kd[ch][k] with k = (kh*3+kw)*32 + c   (c innermost -> constant-stride loads)
// ---------------------------------------------------------------------------
__global__ __launch_bounds__(256) void fold_weights_k(
    const float* __restrict__ w, const int* __restrict__ cn,
    float* __restrict__ kd)
{
  int i = blockIdx.x * 256 + threadIdx.x;          // over COUT*KPAD
  if (i >= NKD) return;
  int ch = i / KPAD;
  int k  = i - ch * KPAD;
  float v = 0.f;
  if (k < KTOT) {
    int khkw = k >> 5;                             // kh*3+kw
    int c    = k & 31;                             // input channel
    int nc   = cn[ch];
    // weights[ch][slot][kh][kw], slots c and c+32 both map to channel c
    const float* wp = w + ch * (MAXCN * 9) + c * 9 + khkw;
    if (c      < nc) v += wp[0];
    if (c + 32 < nc) v += wp[32 * 9];
  }
  kd[i] = v;
}

// ---------------------------------------------------------------------------
// Implicit-GEMM conv via V_WMMA_F32_16X16X4_F32 (full fp32 precision).
// Per wave: 16 flat output pixels (N) x 64 output channels (M, 4 frags),
// K = 288 reduction in 72 WMMA steps of K=4.
// Folded weights staged into LDS by the Tensor Data Mover (TENSORcnt).
// ---------------------------------------------------------------------------
__global__ __launch_bounds__(256) void conv_wmma_k(
    const float* __restrict__ x, const float* __restrict__ kd,
    const float* __restrict__ bias, float* __restrict__ out)
{
  extern __shared__ float kds[];                   // [COUT][KPAD] = 73 KB

#ifdef HAVE_TDM
  if (threadIdx.x < 32) {                          // wave 0 drives the TDM DMA
    unsigned long long ga = (unsigned long long)(const void*)kd;
    unsigned ldsa = (unsigned)(size_t)(void*)kds;  // LDS byte offset (aperture low bits)
    v4u g0;
    g0.x = 1u;                                     // count=1, user descriptor
    g0.y = ldsa;                                   // lds_addr
    g0.z = (unsigned)ga;                           // global_addr[31:0]
    g0.w = ((unsigned)(ga >> 32) & 0x01FFFFFFu)    // global_addr[56:32]
           | (2u << 30);                           // type=2 ("image")
    v8i g1;
    g1[0] = 0x00020000;                            // data_size=2 (4 bytes)
    g1[1] = (int)((NKD & 0xFFFF) << 16);           // tensor_dim0[15:0]
    g1[2] = (int)((NKD >> 16) | (1u << 16));       // tensor_dim0[31:16] | tensor_dim1=1
    g1[3] = (int)((NKD & 0xFFFF) << 16);           // tile_dim0 = NKD
    g1[4] = 1;                                     // tile_dim1=1, tile_dim2=0
    g1[5] = (int)NKD;                              // tensor_dim0_stride[31:0]
    g1[6] = 0;                                     // stride hi | dim1_stride lo
    g1[7] = 0;
    v4i gz4 = {0, 0, 0, 0};                        // groups 2/3 unused (<=2D)
    v8i gz8 = {0, 0, 0, 0, 0, 0, 0, 0};
    __builtin_amdgcn_tensor_load_to_lds(g0, g1, gz4, gz4, gz8, 0);
    __builtin_amdgcn_s_wait_tensorcnt(0);
  }
#else
  {
    const float4* s = (const float4*)kd;
    float4*       d = (float4*)kds;
    for (int i = threadIdx.x; i < NKD / 4; i += 256) d[i] = s[i];
  }
#endif
  __syncthreads();

  const int lane = threadIdx.x & 31;
  const int wid  = threadIdx.x >> 5;
  const int half = lane >> 4;                      // half-wave select
  const int l16  = lane & 15;

  // this lane's pixel column (B-matrix N index / D-matrix N index)
  const int p  = (blockIdx.x * 8 + wid) * 16 + l16;
  const int b  = p / HOWO;
  const int r  = p - b * HOWO;
  const int ho = r / WOo;
  const int wo = r - ho * WOo;

  const int kdel = half * 2;                       // K offset of this half-wave

  // x base: channel = kdel folded in; per-step offset c0*HW + kh*W + kw is imm
  const float* pb = x + ((b * CIN + kdel) * Hh + ho) * Ww + wo;

  // A-fragment bases in LDS (row = output channel, contiguous K-pair -> b64)
  const float* a0 = kds + ( 0 + l16) * KPAD + kdel;
  const float* a1 = kds + (16 + l16) * KPAD + kdel;
  const float* a2 = kds + (32 + l16) * KPAD + kdel;
  const float* a3 = kds + (48 + l16) * KPAD + kdel;

  v8f acc0 = {}, acc1 = {}, acc2 = {}, acc3 = {};

  #pragma unroll 1
  for (int kh = 0; kh < 3; ++kh) {
    #pragma unroll 1
    for (int kw = 0; kw < 3; ++kw) {
      const float* px = pb + kh * Ww + kw;
      const int    kb = (kh * 3 + kw) * 32;
      const float* b0 = a0 + kb;
      const float* b1 = a1 + kb;
      const float* b2 = a2 + kb;
      const float* b3 = a3 + kb;
      #pragma unroll
      for (int s = 0; s < 8; ++s) {               // channel groups of 4
        v2f bf;                                    // B: x values, K pair
        bf.x = px[s * 4 * HW];
        bf.y = px[s * 4 * HW + HW];
        v2f f0 = *(const v2f*)(b0 + s * 4);        // A: weight K pairs
        v2f f1 = *(const v2f*)(b1 + s * 4);
        v2f f2 = *(const v2f*)(b2 + s * 4);
        v2f f3 = *(const v2f*)(b3 + s * 4);
        acc0 = __builtin_amdgcn_wmma_f32_16x16x4_f32(false, f0, false, bf,
                                                     (short)0, acc0, false, false);
        acc1 = __builtin_amdgcn_wmma_f32_16x16x4_f32(false, f1, false, bf,
                                                     (short)0, acc1, false, false);
        acc2 = __builtin_amdgcn_wmma_f32_16x16x4_f32(false, f2, false, bf,
                                                     (short)0, acc2, false, false);
        acc3 = __builtin_amdgcn_wmma_f32_16x16x4_f32(false, f3, false, bf,
                                                     (short)0, acc3, false, false);
      }
    }
  }

  // D layout: lane -> N=l16 pixel, VGPR v -> M = v + 8*half. ch = 16*t + M.
  // Output is write-once/never-reread (124 MB): stream past L2 with NT stores
  // so x (64 MB, high reuse) and kdense stay L2-resident.
  const float* pbia = bias + (half * 8) * HOWO + ho * WOo + wo;
  float*       pout = out  + (b * COUT + half * 8) * HOWO + ho * WOo + wo;
  #pragma unroll
  for (int v = 0; v < 8; ++v) {
    __builtin_nontemporal_store(acc0[v] + pbia[(0 * 16 + v) * HOWO],
                                &pout[(0 * 16 + v) * HOWO]);
    __builtin_nontemporal_store(acc1[v] + pbia[(1 * 16 + v) * HOWO],
                                &pout[(1 * 16 + v) * HOWO]);
    __builtin_nontemporal_store(acc2[v] + pbia[(2 * 16 + v) * HOWO],
                                &pout[(2 * 16 + v) * HOWO]);
    __builtin_nontemporal_store(acc3[v] + pbia[(3 * 16 + v) * HOWO],
                                &pout[(3 * 16 + v) * HOWO]);
  }
}

// ---------------------------------------------------------------------------
extern "C" void kernel_launch(void* const* d_in, const int* in_sizes, int n_in,
                              void* d_out, int out_size, void* d_ws, size_t ws_size,
                              hipStream_t stream) {
  (void)in_sizes; (void)n_in; (void)out_size; (void)ws_size;
  const float* x    = (const float*)d_in[0];
  const float* w    = (const float*)d_in[1];
  const float* bias = (const float*)d_in[2];
  const int*   cn   = (const int*)d_in[3];
  float* kd = (float*)d_ws;                        // 74,752 B scratch

  fold_weights_k<<<(NKD + 255) / 256, 256, 0, stream>>>(w, cn, kd);

  const int n_pixels = 32 * HOWO;                  // 508,032 = 16 * 31,752
  const int n_blocks = n_pixels / (16 * 8);        // 3,969 blocks x 8 waves
  const size_t lds_bytes = (size_t)NKD * sizeof(float);
  conv_wmma_k<<<n_blocks, 256, lds_bytes, stream>>>(x, kd, bias, (float*)d_out);
}